// GCN_23673859735792
// MI455X (gfx1250) — compile-verified
//
#include <hip/hip_runtime.h>

typedef __attribute__((ext_vector_type(2))) float v2f;
typedef __attribute__((ext_vector_type(8))) float v8f;

#define FIN 128

// ======================= CSR construction ==========================
__global__ void k_zero_int(int* __restrict__ p, int n) {
  int i = blockIdx.x * blockDim.x + threadIdx.x;
  if (i < n) p[i] = 0;
}

__global__ void k_count(const long long* __restrict__ dst, int* __restrict__ cnt,
                        long long E) {
  long long e = (long long)blockIdx.x * blockDim.x + threadIdx.x;
  if (e < E) atomicAdd(&cnt[(int)dst[e]], 1);
}

// dinv[i] = rsqrt(indeg + 1)  (self-loop included, always >= 1)
__global__ void k_dinv(const int* __restrict__ cnt, float* __restrict__ dinv, int n) {
  int i = blockIdx.x * blockDim.x + threadIdx.x;
  if (i < n) dinv[i] = rsqrtf((float)(cnt[i] + 1));
}

// Single-workgroup exclusive scan (1024 threads, LDS Hillis-Steele per chunk).
__global__ void k_scan(const int* __restrict__ cnt, int* __restrict__ rowstart, int n) {
  __shared__ int buf[1024];
  __shared__ int carry;
  if (threadIdx.x == 0) carry = 0;
  __syncthreads();
  for (int base = 0; base < n; base += 1024) {
    int i = base + (int)threadIdx.x;
    int v = (i < n) ? cnt[i] : 0;
    buf[threadIdx.x] = v;
    __syncthreads();
    #pragma unroll
    for (int off = 1; off < 1024; off <<= 1) {
      int t = (threadIdx.x >= (unsigned)off) ? buf[threadIdx.x - off] : 0;
      __syncthreads();
      buf[threadIdx.x] += t;
      __syncthreads();
    }
    int incl = buf[threadIdx.x];
    if (i < n) rowstart[i] = incl - v + carry;   // exclusive + running offset
    __syncthreads();
    if (threadIdx.x == 1023) carry += incl;      // chunk total
    __syncthreads();
  }
  if (threadIdx.x == 0) rowstart[n] = carry;
}

__global__ void k_fill(const long long* __restrict__ src, const long long* __restrict__ dst,
                       const int* __restrict__ rowstart, int* __restrict__ fill,
                       int* __restrict__ csr, long long E) {
  long long e = (long long)blockIdx.x * blockDim.x + threadIdx.x;
  if (e < E) {
    int d = (int)dst[e];
    int p = atomicAdd(&fill[d], 1);
    csr[rowstart[d] + p] = (int)src[e];
  }
}

// ============ fp32 WMMA GEMM: H[N,FOUT] = act(X[N,128]) @ W[FOUT,128]^T ======
// One wave computes one 16x16 tile via v_wmma_f32_16x16x4_f32 (full fp32).
template <int FOUT, bool RELU_IN>
__global__ void k_gemm_wmma(const float* __restrict__ X,
                            const float* __restrict__ W,
                            float* __restrict__ H, int n) {
  constexpr int NT = FOUT / 16;
  const int wid  = (blockIdx.x * blockDim.x + threadIdx.x) >> 5;
  const int lane = threadIdx.x & 31;
  const int tm = wid / NT;
  const int tn = wid % NT;
  if (tm * 16 >= n) return;       // wave-uniform: EXEC stays all-1s for WMMA

  const int half = lane >> 4;     // 0: K={0,1}, 1: K={2,3}
  const int l16  = lane & 15;

  const float* xa = X + (size_t)(tm * 16 + l16) * FIN + half * 2;  // A 16x4
  const float* wb = W + (size_t)(tn * 16 + l16) * FIN + half * 2;  // B 4x16 (=W^T)

  v8f acc = {};
#pragma unroll
  for (int k = 0; k < FIN; k += 4) {
    float a0 = xa[k], a1 = xa[k + 1];
    if (RELU_IN) { a0 = fmaxf(a0, 0.0f); a1 = fmaxf(a1, 0.0f); }
    v2f a, b;
    a[0] = a0;      a[1] = a1;
    b[0] = wb[k];   b[1] = wb[k + 1];
    acc = __builtin_amdgcn_wmma_f32_16x16x4_f32(
        false, a, false, b, (short)0, acc, false, false);
  }

  float* out = H + (size_t)(tm * 16 + half * 8) * FOUT + tn * 16 + l16;
#pragma unroll
  for (int r = 0; r < 8; ++r) out[(size_t)r * FOUT] = acc[r];
}

// ========== pull aggregation: out[d] = sum_{s in N(d)} h[s]*dinv[s]*dinv[d]
//                                      + h[d]*dinv[d]^2 + bias  (no atomics) ==
template <int FOUT>
__global__ void k_gather(const int* __restrict__ csr, const int* __restrict__ rowstart,
                         const int* __restrict__ cnt, const float* __restrict__ dinv,
                         const float* __restrict__ h, const float* __restrict__ bias,
                         float* __restrict__ out, int n) {
  constexpr int PER = FOUT / 32;   // floats per lane (4 for F=128, 2 for F=64)
  const int node = (blockIdx.x * blockDim.x + threadIdx.x) >> 5;  // wave-uniform
  const int lane = threadIdx.x & 31;
  if (node >= n) return;

  const int f  = lane * PER;
  const float dn = dinv[node];     // scalar load (uniform)
  const float w0 = dn * dn;

  float acc[PER];
  const float* hd = h + (size_t)node * FOUT + f;
#pragma unroll
  for (int j = 0; j < PER; ++j) acc[j] = hd[j] * w0 + bias[f + j];

  const int beg = rowstart[node];  // uniform -> scalar load
  const int m   = cnt[node];       // uniform -> scalar load
  for (int t = 0; t < m; ++t) {
    int s = csr[beg + t];          // uniform -> scalar load
    if (t + 1 < m) {               // prefetch next source row (global_prefetch_b8)
      int sn = csr[beg + t + 1];
      __builtin_prefetch(h + (size_t)sn * FOUT + f, 0, 3);
    }
    float w = dinv[s] * dn;
    const float* hs = h + (size_t)s * FOUT + f;   // coalesced row read (L2-resident)
#pragma unroll
    for (int j = 0; j < PER; ++j) acc[j] = fmaf(hs[j], w, acc[j]);
  }

  float* o = out + (size_t)node * FOUT + f;
#pragma unroll
  for (int j = 0; j < PER; ++j) o[j] = acc[j];
}

// =============================== driver ====================================
extern "C" void kernel_launch(void* const* d_in, const int* in_sizes, int n_in,
                              void* d_out, int out_size, void* d_ws, size_t ws_size,
                              hipStream_t stream) {
  const float*     x  = (const float*)d_in[0];
  const long long* ei = (const long long*)d_in[1];   // int64 [2, E]
  const float*     W1 = (const float*)d_in[2];
  const float*     b1 = (const float*)d_in[3];
  const float*     W2 = (const float*)d_in[4];
  const float*     b2 = (const float*)d_in[5];

  const int       N = in_sizes[0] / 128;             // 100000 (multiple of 16)
  const long long E = (long long)in_sizes[1] / 2;
  const long long* src = ei;
  const long long* dst = ei + E;

  // -------- workspace layout (256-element aligned regions) --------
  size_t NA = (size_t)((N + 256) & ~255);            // room for N+1 ints
  int*   cnt      = (int*)d_ws;
  int*   fill     = cnt + NA;
  int*   rowstart = fill + NA;                       // N+1 entries
  int*   csr      = rowstart + NA;
  size_t EA = (size_t)((E + 255) & ~255ll);
  float* dinv = (float*)(csr + EA);
  float* h1   = dinv + NA;
  float* agg1 = h1 + (size_t)N * 128;
  float* h2   = agg1 + (size_t)N * 128;

  const int B = 256;
  const int gN  = (N + B - 1) / B;
  const int gE  = (int)((E + B - 1) / B);

  // -------- CSR + normalization (shared by both layers) --------
  k_zero_int<<<gN, B, 0, stream>>>(cnt, N);
  k_count<<<gE, B, 0, stream>>>(dst, cnt, E);
  k_dinv<<<gN, B, 0, stream>>>(cnt, dinv, N);
  k_scan<<<1, 1024, 0, stream>>>(cnt, rowstart, N);
  k_zero_int<<<gN, B, 0, stream>>>(fill, N);
  k_fill<<<gE, B, 0, stream>>>(src, dst, rowstart, fill, csr, E);

  // -------- layer 1: h1 = x @ W1^T ; agg1 = Ahat h1 + b1 --------
  {
    long long waves = (long long)(N / 16) * (128 / 16);
    k_gemm_wmma<128, false>
        <<<(int)((waves * 32 + B - 1) / B), B, 0, stream>>>(x, W1, h1, N);
    long long gw = ((long long)N * 32 + B - 1) / B;
    k_gather<128><<<(int)gw, B, 0, stream>>>(csr, rowstart, cnt, dinv, h1, b1, agg1, N);
  }

  // -------- layer 2: h2 = relu(agg1) @ W2^T ; out = Ahat h2 + b2 --------
  {
    long long waves = (long long)(N / 16) * (64 / 16);
    k_gemm_wmma<64, true>
        <<<(int)((waves * 32 + B - 1) / B), B, 0, stream>>>(agg1, W2, h2, N);
    long long gw = ((long long)N * 32 + B - 1) / B;
    k_gather<64><<<(int)gw, B, 0, stream>>>(csr, rowstart, cnt, dinv, h2, b2,
                                            (float*)d_out, N);
  }
}